// TinyYOLAF_53060025975361
// MI455X (gfx1250) — compile-verified
//
#include <hip/hip_runtime.h>
#include <cstdint>

// ---------------- problem constants ----------------
#define N_TOT   100800      // HW * A = 33600 * 3
#define KTOP    4096
#define HB      4096        // histogram bins (12 bits)
#define CAP     8192        // candidate cap (>= KTOP + tie slack)
#define CONF_T  0.01f
#define NMS_T   0.3f
#define EPS_F   1e-28f

// ---------------- workspace layout (bytes) ----------------
#define OFF_SCORES 0ull
#define OFF_BOXES  403456ull                     // after N*4 scores, 256B aligned
#define OFF_HIST1  (OFF_BOXES + 1612800ull)      // after N*16 boxes
#define OFF_HIST2  (OFF_HIST1 + 16384ull)
#define OFF_SEL    (OFF_HIST2 + 16384ull)        // sel[0]=T, sel[1]=above, sel[2]=T2full, sel[3]=candCnt
#define OFF_CAND   (OFF_SEL   + 256ull)
#define OFF_TVALS  (OFF_CAND  + 65536ull)        // CAP * 8
#define OFF_TIDX   (OFF_TVALS + 16384ull)
#define OFF_CBOX   (OFF_TIDX  + 16384ull)
#define OFF_MASK   (OFF_CBOX  + 65536ull)        // KTOP * 16
#define OFF_KEEP   (OFF_MASK  + 2097152ull)      // KTOP * 128 * 4
// total ~4.31 MB

// ---------------- CDNA5 async global->LDS path ----------------
#if defined(__has_builtin)
# if __has_builtin(__builtin_amdgcn_global_load_async_to_lds_b128)
#  define USE_ASYNC_LDS 1
# endif
# if __has_builtin(__builtin_amdgcn_s_wait_asynccnt)
#  define HAVE_WAIT_ASYNC 1
# endif
#endif

#if defined(USE_ASYNC_LDS)
// Builtin signature (from clang diagnostic): param0 = 'int __vector(4) __device__ *'
// i.e. non-const gcc-vector int4 pointer in addrspace(1); param1 = addrspace(3) twin.
typedef int b128_t __attribute__((vector_size(16)));
typedef __attribute__((address_space(1))) b128_t* gbl_b128_p;
typedef __attribute__((address_space(3))) b128_t* lds_b128_p;
// LDS generic pointer low 32 bits == LDS byte offset (CDNA5 aperture truncation rule);
// global generic pointer value == AS(1) value. Integer round-trip avoids addrspacecast issues.
#define TO_GBL_B128(p) ((gbl_b128_p)(uintptr_t)(p))
#define TO_LDS_B128(p) ((lds_b128_p)(uint32_t)(uintptr_t)(p))
#endif

__device__ __forceinline__ void wait_async_zero() {
#if defined(HAVE_WAIT_ASYNC)
  __builtin_amdgcn_s_wait_asynccnt(0);
#else
  asm volatile("s_wait_asynccnt 0" ::: "memory");
#endif
}

// ---------------- anchors ----------------
__constant__ float c_anchor[9][2] = {
  {10.f, 13.f}, {16.f, 30.f}, {33.f, 23.f},
  {30.f, 61.f}, {62.f, 45.f}, {59.f, 119.f},
  {116.f, 90.f}, {156.f, 198.f}, {373.f, 326.f}};

__device__ __forceinline__ float sigmoidf_(float x) { return 1.f / (1.f + expf(-x)); }
__device__ __forceinline__ float clamp01_(float x) { return fminf(fmaxf(x, 0.f), 1.f); }

// ---------------- kernels ----------------
__global__ void zero_kernel(unsigned* p, int n) {
  int i = blockIdx.x * blockDim.x + threadIdx.x;
  if (i < n) p[i] = 0u;
}

// scores (thresholded sigmoid) + decoded boxes (image 0) + 12-bit histogram
__global__ void decode_kernel(const float* __restrict__ conf,
                              const float* __restrict__ txty,
                              float* __restrict__ scores,
                              float* __restrict__ boxes,
                              unsigned* __restrict__ hist1) {
  int n = blockIdx.x * blockDim.x + threadIdx.x;
  if (n >= N_TOT) return;

  float s = sigmoidf_(conf[n]);
  s = (s >= CONF_T) ? s : 0.f;
  scores[n] = s;
  atomicAdd(&hist1[__float_as_uint(s) >> 20], 1u);

  int hw = n / 3, a = n - hw * 3;
  int lvl, gx, gy; float st;
  if (hw < 25600)       { lvl = 0; st = 8.f;  gx = hw % 160;            gy = hw / 160; }
  else if (hw < 32000)  { int l = hw - 25600; lvl = 1; st = 16.f; gx = l % 80; gy = l / 80; }
  else                  { int l = hw - 32000; lvl = 2; st = 32.f; gx = l % 40; gy = l / 40; }

  const float* t = txty + (size_t)n * 4;
  float cx = (sigmoidf_(t[0]) + (float)gx) * st;
  float cy = (sigmoidf_(t[1]) + (float)gy) * st;
  float bw = expf(t[2]) * c_anchor[lvl * 3 + a][0];
  float bh = expf(t[3]) * c_anchor[lvl * 3 + a][1];
  const float inv = 1.f / 1280.f;
  float* o = boxes + (size_t)n * 4;
  o[0] = clamp01_((cx - bw * 0.5f) * inv);
  o[1] = clamp01_((cy - bh * 0.5f) * inv);
  o[2] = clamp01_((cx + bw * 0.5f) * inv);
  o[3] = clamp01_((cy + bh * 0.5f) * inv);
}

__global__ void scan1_kernel(const unsigned* __restrict__ hist1, int* __restrict__ sel) {
  if (threadIdx.x != 0 || blockIdx.x != 0) return;
  unsigned cum = 0;
  for (int b = HB - 1; b >= 0; --b) {
    unsigned c = hist1[b];
    if (cum + c >= (unsigned)KTOP) { sel[0] = b; sel[1] = (int)cum; return; }
    cum += c;
  }
  sel[0] = 0; sel[1] = (int)cum;   // unreachable (total = N_TOT >= KTOP)
}

__global__ void hist2_kernel(const float* __restrict__ scores,
                             const int* __restrict__ sel,
                             unsigned* __restrict__ hist2) {
  int n = blockIdx.x * blockDim.x + threadIdx.x;
  if (n >= N_TOT) return;
  unsigned bits = __float_as_uint(scores[n]);
  if ((int)(bits >> 20) == sel[0]) atomicAdd(&hist2[(bits >> 8) & 0xFFFu], 1u);
}

__global__ void scan2_kernel(const unsigned* __restrict__ hist2, int* __restrict__ sel) {
  if (threadIdx.x != 0 || blockIdx.x != 0) return;
  unsigned cum = (unsigned)sel[1];
  int T2 = 0;
  for (int b = HB - 1; b >= 0; --b) {
    unsigned c = hist2[b];
    if (cum + c >= (unsigned)KTOP) { T2 = b; break; }
    cum += c;
  }
  sel[2] = (sel[0] << 12) | T2;    // 24-bit threshold prefix
}

__global__ void compact_kernel(const float* __restrict__ scores,
                               int* __restrict__ sel,
                               unsigned long long* __restrict__ cand) {
  int n = blockIdx.x * blockDim.x + threadIdx.x;
  if (n >= N_TOT) return;
  unsigned bits = __float_as_uint(scores[n]);
  if (bits != 0u && (bits >> 8) >= (unsigned)sel[2]) {
    unsigned p = atomicAdd((unsigned*)&sel[3], 1u);
    if (p < (unsigned)CAP)
      cand[p] = ((unsigned long long)bits << 32) | (unsigned)(~(unsigned)n);
  }
}

// one-workgroup bitonic sort of CAP 64-bit keys in LDS (64 KB), descending;
// emit top-K values/indices and gather candidate boxes.
__global__ __launch_bounds__(1024) void sort_kernel(const int* __restrict__ sel,
                                                    const unsigned long long* __restrict__ cand,
                                                    const float* __restrict__ boxes,
                                                    float* __restrict__ tvals,
                                                    unsigned* __restrict__ tidx,
                                                    float4* __restrict__ cbox) {
  __shared__ unsigned long long sk[CAP];
  int tid = threadIdx.x;
  unsigned cnt = (unsigned)sel[3]; if (cnt > (unsigned)CAP) cnt = (unsigned)CAP;
  for (int i = tid; i < CAP; i += 1024) sk[i] = (i < (int)cnt) ? cand[i] : 0ull;
  __syncthreads();

  for (int k = 2; k <= CAP; k <<= 1) {
    for (int j = k >> 1; j > 0; j >>= 1) {
      for (int i = tid; i < CAP; i += 1024) {
        int p = i ^ j;
        if (p > i) {
          unsigned long long a = sk[i], b = sk[p];
          bool up = (i & k) == 0;                 // descending overall
          if (up ? (a < b) : (a > b)) { sk[i] = b; sk[p] = a; }
        }
      }
      __syncthreads();
    }
  }

  for (int i = tid; i < KTOP; i += 1024) {
    unsigned long long key = sk[i];
    unsigned vb = (unsigned)(key >> 32);
    unsigned idx = ~(unsigned)(key & 0xFFFFFFFFull);
    if (vb == 0u || idx >= (unsigned)N_TOT) idx = 0u;   // zero-pad entries: boxes are masked out anyway
    tvals[i] = __uint_as_float(vb);
    tidx[i] = idx;
    const float* bp = boxes + (size_t)idx * 4;
    cbox[i] = make_float4(bp[0], bp[1], bp[2], bp[3]);
  }
}

// suppression bitmask: mask[i*128+w] bit b  <=>  iou(i, w*32+b) > NMS_T  &&  j > i
// boxes staged into LDS via CDNA5 async global->LDS loads.
__global__ __launch_bounds__(1024) void mask_kernel(const float4* __restrict__ cbox,
                                                    unsigned* __restrict__ mask) {
  __shared__ float4 sb[KTOP];  // 64 KB
  int tid = threadIdx.x;
#if defined(USE_ASYNC_LDS)
  for (int q = 0; q < 4; ++q) {
    int idx = tid + 1024 * q;
    __builtin_amdgcn_global_load_async_to_lds_b128(TO_GBL_B128(&cbox[idx]),
                                                   TO_LDS_B128(&sb[idx]), 0, 0);
  }
  wait_async_zero();
  __syncthreads();
#else
  for (int q = 0; q < 4; ++q) { int idx = tid + 1024 * q; sb[idx] = cbox[idx]; }
  __syncthreads();
#endif

  int i = blockIdx.x * 32 + (tid >> 5);     // 128 blocks x 32 rows
  int lane = tid & 31;
  float4 bi = sb[i];
  float areai = (bi.z - bi.x) * (bi.w - bi.y);

  for (int q = 0; q < 4; ++q) {
    int w = lane + 32 * q;                  // word 0..127
    int jbase = w * 32;
    unsigned m = 0;
    for (int b = 0; b < 32; ++b) {
      int bb = (b + lane) & 31;             // rotate to kill LDS bank conflicts
      int j = jbase + bb;
      float4 bj = sb[j];
      float xx1 = fmaxf(bi.x, bj.x), yy1 = fmaxf(bi.y, bj.y);
      float xx2 = fminf(bi.z, bj.z), yy2 = fminf(bi.w, bj.w);
      float inter = fmaxf(EPS_F, xx2 - xx1) * fmaxf(EPS_F, yy2 - yy1);
      float areaj = (bj.z - bj.x) * (bj.w - bj.y);
      float iou = inter / (areai + areaj - inter);
      if (iou > NMS_T && j > i) m |= (1u << bb);
    }
    mask[(size_t)i * 128 + w] = m;
  }
}

// single wave32: keep mask (4096 bits) lives in 4 VGPR words per lane; no barriers.
__global__ void nms_seq_kernel(const unsigned* __restrict__ mask,
                               const float* __restrict__ tvals,
                               unsigned* __restrict__ keepOut) {
  int lane = threadIdx.x;                   // 32 threads
  unsigned k0 = 0, k1 = 0, k2 = 0, k3 = 0;
  for (int r = 0; r < 4; ++r) {
    int w = lane + 32 * r;
    unsigned m = 0;
    for (int b = 0; b < 32; ++b)
      if (tvals[w * 32 + b] >= CONF_T) m |= (1u << b);
    if (r == 0) k0 = m; else if (r == 1) k1 = m; else if (r == 2) k2 = m; else k3 = m;
  }

  for (int i = 0; i < KTOP; ++i) {
    int w = i >> 5;
    int owner = w & 31;
    unsigned word;
    switch (i >> 10) {                      // wave-uniform
      case 0:  word = __shfl(k0, owner, 32); break;
      case 1:  word = __shfl(k1, owner, 32); break;
      case 2:  word = __shfl(k2, owner, 32); break;
      default: word = __shfl(k3, owner, 32); break;
    }
    if ((word >> (i & 31)) & 1u) {          // wave-uniform branch
      const unsigned* row = mask + (size_t)i * 128;
      k0 &= ~row[lane];
      k1 &= ~row[lane + 32];
      k2 &= ~row[lane + 64];
      k3 &= ~row[lane + 96];
    }
  }
  keepOut[lane] = k0; keepOut[lane + 32] = k1;
  keepOut[lane + 64] = k2; keepOut[lane + 96] = k3;
}

__global__ void final_kernel(const float4* __restrict__ cbox,
                             const float* __restrict__ tvals,
                             const unsigned* __restrict__ keep,
                             float* __restrict__ out) {
  int i = blockIdx.x * blockDim.x + threadIdx.x;
  if (i >= KTOP) return;
  float f = ((keep[i >> 5] >> (i & 31)) & 1u) ? 1.f : 0.f;
  float4 b = cbox[i];
  out[i * 4 + 0] = b.x * f;
  out[i * 4 + 1] = b.y * f;
  out[i * 4 + 2] = b.z * f;
  out[i * 4 + 3] = b.w * f;
  out[KTOP * 4 + i] = tvals[i] * f;         // scores after the [K,4] boxes
}

// ---------------- launcher ----------------
extern "C" void kernel_launch(void* const* d_in, const int* in_sizes, int n_in,
                              void* d_out, int out_size, void* d_ws, size_t ws_size,
                              hipStream_t stream) {
  (void)in_sizes; (void)n_in; (void)out_size; (void)ws_size;
  const float* conf = (const float*)d_in[0];   // [8, N, 1]; image 0 = first N
  const float* txty = (const float*)d_in[1];   // [8, N, 4]; image 0 = first 4N
  float* out = (float*)d_out;                  // 4096*4 boxes + 4096 scores

  char* ws = (char*)d_ws;
  float*              scores = (float*)(ws + OFF_SCORES);
  float*              boxes  = (float*)(ws + OFF_BOXES);
  unsigned*           hist1  = (unsigned*)(ws + OFF_HIST1);
  int*                sel    = (int*)(ws + OFF_SEL);
  unsigned*           hist2  = (unsigned*)(ws + OFF_HIST2);
  unsigned long long* cand   = (unsigned long long*)(ws + OFF_CAND);
  float*              tvals  = (float*)(ws + OFF_TVALS);
  unsigned*           tidx   = (unsigned*)(ws + OFF_TIDX);
  float4*             cbox   = (float4*)(ws + OFF_CBOX);
  unsigned*           maskm  = (unsigned*)(ws + OFF_MASK);
  unsigned*           keep   = (unsigned*)(ws + OFF_KEEP);

  const int nblk = (N_TOT + 255) / 256;

  // hist1 + hist2 + sel are contiguous: (16384+16384+256)/4 = 8256 words
  zero_kernel<<<(8256 + 255) / 256, 256, 0, stream>>>(hist1, 8256);
  decode_kernel<<<nblk, 256, 0, stream>>>(conf, txty, scores, boxes, hist1);
  scan1_kernel<<<1, 1, 0, stream>>>(hist1, sel);
  hist2_kernel<<<nblk, 256, 0, stream>>>(scores, sel, hist2);
  scan2_kernel<<<1, 1, 0, stream>>>(hist2, sel);
  compact_kernel<<<nblk, 256, 0, stream>>>(scores, sel, cand);
  sort_kernel<<<1, 1024, 0, stream>>>(sel, cand, boxes, tvals, tidx, cbox);
  mask_kernel<<<128, 1024, 0, stream>>>(cbox, maskm);
  nms_seq_kernel<<<1, 32, 0, stream>>>(maskm, tvals, keep);
  final_kernel<<<(KTOP + 255) / 256, 256, 0, stream>>>(cbox, tvals, keep, out);
}